// Top_k_6631429505461
// MI455X (gfx1250) — compile-verified
//
#include <hip/hip_runtime.h>
#include <math.h>

typedef float v2f __attribute__((ext_vector_type(2)));
typedef float v8f __attribute__((ext_vector_type(8)));

#define B_CONST 16
#define D_CONST 256
#define NPT 16            // N / 1024 with N = 16384
#define NEG_BIG (-9.0e15f)
#define LOG_EPS (-27.631021f)   // logf(1e-12f)

// ---------------------------------------------------------------------------
// Kernel 1: u[b,:] = W @ (embed[b,0,:] @ W)   via f32 WMMA, one workgroup.
//   Step A: q = e0 @ W          (16 x 256) = (16 x 256)(256 x 256)
//   Step B: u = q  @ W^T        (16 x 256)
// 16 waves; wave j owns output columns [16j, 16j+16).
// ---------------------------------------------------------------------------
__global__ __launch_bounds__(512) void qu_wmma_kernel(
    const float* __restrict__ embed,   // [B, N, D]
    const float* __restrict__ W,       // [D, D]
    float* __restrict__ u_out,         // [B_CONST, D_CONST]
    int N) {
  __shared__ float q_lds[B_CONST * D_CONST];   // 16 KB

  const int t    = threadIdx.x;
  const int lane = t & 31;
  const int wave = t >> 5;        // 0..15
  const int half = lane >> 4;     // 0 (K=0,1) or 1 (K=2,3)
  const int l16  = lane & 15;
  const int colBase = wave * 16;

  // ---- Step A: q = e0 @ W ----
  v8f c = {0.f, 0.f, 0.f, 0.f, 0.f, 0.f, 0.f, 0.f};
  for (int kk = 0; kk < D_CONST; kk += 4) {
    const int ka = kk + half * 2;
    v2f a, b;
    // A fragment: A[M = l16][ka], A[M = l16][ka+1]  (e0[b, d] = embed[b*N*D + d])
    const float* e0row = embed + (size_t)l16 * (size_t)N * D_CONST;
    a.x = e0row[ka];
    a.y = e0row[ka + 1];
    // B fragment: B[k][n] = W[k, colBase + n]
    b.x = W[(size_t)ka * D_CONST + colBase + l16];
    b.y = W[(size_t)(ka + 1) * D_CONST + colBase + l16];
    c = __builtin_amdgcn_wmma_f32_16x16x4_f32(false, a, false, b,
                                              (short)0, c, false, false);
  }
  // Store q tile to LDS: C VGPR r holds rows r (lanes 0-15) / r+8 (lanes 16-31)
#pragma unroll
  for (int r = 0; r < 8; ++r) {
    const int row = r + half * 8;
    q_lds[row * D_CONST + colBase + l16] = c[r];
  }
  __syncthreads();

  // ---- Step B: u = q @ W^T ----
  v8f c2 = {0.f, 0.f, 0.f, 0.f, 0.f, 0.f, 0.f, 0.f};
  for (int kk = 0; kk < D_CONST; kk += 4) {
    const int ka = kk + half * 2;
    v2f a, b;
    a.x = q_lds[l16 * D_CONST + ka];
    a.y = q_lds[l16 * D_CONST + ka + 1];
    // B'[k][n] = W^T[k, colBase+n] = W[colBase+n, k]
    b.x = W[(size_t)(colBase + l16) * D_CONST + ka];
    b.y = W[(size_t)(colBase + l16) * D_CONST + ka + 1];
    c2 = __builtin_amdgcn_wmma_f32_16x16x4_f32(false, a, false, b,
                                               (short)0, c2, false, false);
  }
#pragma unroll
  for (int r = 0; r < 8; ++r) {
    const int row = r + half * 8;
    u_out[row * D_CONST + colBase + l16] = c2[r];
  }
}

// ---------------------------------------------------------------------------
// Kernel 2: scores[b,n] = mask ? dot(embed[b,n,:], u[b,:]) / sqrt(D) : NEG_BIG
// One wave32 per row; float4 coalesced loads; shfl-xor reduction.
// ---------------------------------------------------------------------------
__global__ __launch_bounds__(256) void scores_kernel(
    const float* __restrict__ embed,   // [B*N, D]
    const int*   __restrict__ m,       // [B*N]
    const float* __restrict__ u,       // [B, D]
    float* __restrict__ scores,        // [B*N]
    int N, float invSqrtD) {
  const int wave = threadIdx.x >> 5;
  const int lane = threadIdx.x & 31;
  const int r = blockIdx.x * 8 + wave;          // global row in [0, B*N)
  const int b = r / N;

  const float* erow = embed + (size_t)r * D_CONST;
  const float* urow = u + (size_t)b * D_CONST;

  float sum = 0.f;
#pragma unroll
  for (int i = 0; i < D_CONST; i += 128) {
    const int d = i + lane * 4;
    const float4 e  = *reinterpret_cast<const float4*>(erow + d);
    const float4 uv = *reinterpret_cast<const float4*>(urow + d);
    sum += e.x * uv.x + e.y * uv.y + e.z * uv.z + e.w * uv.w;
  }
#pragma unroll
  for (int off = 16; off > 0; off >>= 1) sum += __shfl_xor(sum, off, 32);

  if (lane == 0) scores[r] = (m[r] > 0) ? sum * invSqrtD : NEG_BIG;
}

// ---------------------------------------------------------------------------
// Kernel 3: per-batch softmax over N, then tk-iteration iterative top-k.
// One 1024-thread block per batch; each thread owns NPT=16 elements in regs.
// Writes RAW y back into ybuf (thresholding done later by thresh_kernel).
// ---------------------------------------------------------------------------
__device__ __forceinline__ float block_reduce(float v, bool isMax, float* sdata) {
  const int lane = threadIdx.x & 31;
  const int wid  = threadIdx.x >> 5;
#pragma unroll
  for (int off = 16; off > 0; off >>= 1) {
    const float o = __shfl_xor(v, off, 32);
    v = isMax ? fmaxf(v, o) : (v + o);
  }
  if (lane == 0) sdata[wid] = v;
  __syncthreads();
  if (wid == 0) {
    float x = (lane < (int)(blockDim.x >> 5)) ? sdata[lane]
                                              : (isMax ? -INFINITY : 0.0f);
#pragma unroll
    for (int off = 16; off > 0; off >>= 1) {
      const float o = __shfl_xor(x, off, 32);
      x = isMax ? fmaxf(x, o) : (x + o);
    }
    if (lane == 0) sdata[0] = x;
  }
  __syncthreads();
  const float res = sdata[0];
  __syncthreads();   // safe reuse of sdata on next call
  return res;
}

__global__ __launch_bounds__(1024) void topk_kernel(
    float* __restrict__ ybuf,          // in: scores, out: raw y  [B*N]
    const int* __restrict__ tkp,
    int N) {
  __shared__ float sdata[32];
  const int b = blockIdx.x;
  const int t = threadIdx.x;
  const int tk = *tkp;

  float att[NPT], yv[NPT];
  float lmax = -INFINITY;
#pragma unroll
  for (int i = 0; i < NPT; ++i) {
    att[i] = ybuf[(size_t)b * N + i * 1024 + t];
    lmax = fmaxf(lmax, att[i]);
  }
  const float M = block_reduce(lmax, true, sdata);
  float lsum = 0.f;
#pragma unroll
  for (int i = 0; i < NPT; ++i) {
    att[i] = expf(att[i] - M);
    lsum += att[i];
  }
  const float S = block_reduce(lsum, false, sdata);
  const float invS = 1.0f / S;
#pragma unroll
  for (int i = 0; i < NPT; ++i) { att[i] *= invS; yv[i] = 0.f; }

  for (int it = 0; it < tk; ++it) {
    float y1[NPT], wv[NPT];
    float m1 = -INFINITY;
#pragma unroll
    for (int i = 0; i < NPT; ++i) {
      const float w = (yv[i] > 0.9f) ? 0.f : 1.f;
      wv[i] = w;
      const float v = att[i] + ((w == 0.f) ? LOG_EPS : 0.0f);
      y1[i] = v;
      m1 = fmaxf(m1, v);
    }
    const float M1 = block_reduce(m1, true, sdata);
    float s1 = 0.f;
#pragma unroll
    for (int i = 0; i < NPT; ++i) {
      const float denom = wv[i] * 1e-6f + (1.0f - wv[i]) * 0.01f;
      const float e = expf((y1[i] - M1) / denom);
      y1[i] = e;
      s1 += e;
    }
    const float S1 = block_reduce(s1, false, sdata);
    const float inv = 1.0f / (S1 + 1e-12f);
#pragma unroll
    for (int i = 0; i < NPT; ++i) yv[i] += y1[i] * inv * wv[i];
  }

#pragma unroll
  for (int i = 0; i < NPT; ++i)
    ybuf[(size_t)b * N + i * 1024 + t] = yv[i];
}

// ---------------------------------------------------------------------------
// Kernel 4: feat_out = feature * y  (float4 streaming; bandwidth-bound pass)
// ---------------------------------------------------------------------------
__global__ __launch_bounds__(256) void scale_kernel(
    const float4* __restrict__ feat4,
    const float*  __restrict__ y,
    float4* __restrict__ out4) {
  const size_t idx = (size_t)blockIdx.x * 256 + threadIdx.x;
  const float yy = y[idx / (D_CONST / 4)];
  const float4 f = feat4[idx];
  float4 o;
  o.x = f.x * yy; o.y = f.y * yy; o.z = f.z * yy; o.w = f.w * yy;
  out4[idx] = o;
}

// ---------------------------------------------------------------------------
// Kernel 5: y_out = (y < 0.1) ? 0 : y   (in place on the y region)
// ---------------------------------------------------------------------------
__global__ __launch_bounds__(256) void thresh_kernel(float* __restrict__ y, int BN) {
  const int i = blockIdx.x * 256 + threadIdx.x;
  if (i < BN) {
    const float v = y[i];
    y[i] = (v < 0.1f) ? 0.0f : v;
  }
}

// ---------------------------------------------------------------------------
extern "C" void kernel_launch(void* const* d_in, const int* in_sizes, int n_in,
                              void* d_out, int out_size, void* d_ws, size_t ws_size,
                              hipStream_t stream) {
  (void)n_in; (void)out_size; (void)d_ws; (void)ws_size;

  const float* feature = (const float*)d_in[0];   // [B,N,D] f32
  const int*   m       = (const int*)  d_in[1];   // [B,N]   i32
  const float* embed   = (const float*)d_in[2];   // [B,N,D] f32
  const float* W       = (const float*)d_in[3];   // [D,D]   f32
  const int*   tkp     = (const int*)  d_in[4];   // scalar

  const int BN  = in_sizes[1];          // B*N = 262144
  const int BND = in_sizes[0];          // B*N*D
  const int N   = BN / B_CONST;         // 16384

  float* out = (float*)d_out;
  // Scratch layout inside d_out (everything gets overwritten in order):
  //  - u lives in the first 4096 floats of the feat region (kernel 4 overwrites)
  //  - scores / raw y live in the y region (kernel 5 finalizes it)
  float* u_buf = out;
  float* y_buf = out + (size_t)BND;

  qu_wmma_kernel<<<1, 512, 0, stream>>>(embed, W, u_buf, N);

  scores_kernel<<<BN / 8, 256, 0, stream>>>(embed, m, u_buf, y_buf, N,
                                            1.0f / sqrtf((float)D_CONST));

  topk_kernel<<<B_CONST, 1024, 0, stream>>>(y_buf, tkp, N);

  scale_kernel<<<(BND / 4) / 256, 256, 0, stream>>>(
      (const float4*)feature, y_buf, (float4*)d_out);

  thresh_kernel<<<(BN + 255) / 256, 256, 0, stream>>>(y_buf, BN);
}